// DigitConvolutionalModel_38474317038232
// MI455X (gfx1250) — compile-verified
//
#include <hip/hip_runtime.h>

typedef __attribute__((ext_vector_type(2))) float v2f;
typedef __attribute__((ext_vector_type(8))) float v8f;

#define TS   32    // samples per block = two WMMA M tiles (B-fragment reuse x2)
#define HSTR 680   // padded LDS stride for conv output row (676 -> 680)
#define IMG  784
#define CONV 676
#define H1   128
#define H2   10

__global__ __launch_bounds__(256) void fused_digit_mlp(
    const float* __restrict__ x,
    const float* __restrict__ conv_w,
    const float* __restrict__ fc1_w,
    const float* __restrict__ fc1_b,
    const float* __restrict__ fc2_w,
    const float* __restrict__ fc2_b,
    float* __restrict__ out)
{
    __shared__ float h[TS * HSTR];          // conv output, 32 x 676 (padded)
    float* h2 = (float*)h;                  // fc1+relu output (32x128) aliases h after barrier

    const int tid = threadIdx.x;
    const int s0  = blockIdx.x * TS;

    // 3x3 conv weights -> registers (uniform, L2-hot)
    float w[9];
    #pragma unroll
    for (int i = 0; i < 9; ++i) w[i] = conv_w[i];

    // ---------------- conv stage: 32 samples x 676 outputs ----------------
    const float* xb = x + (size_t)s0 * IMG;
    for (int t = tid; t < TS * CONV; t += 256) {
        int s  = t / CONV;
        int p  = t - s * CONV;
        int oy = p / 26;
        int ox = p - oy * 26;
        const float* px = xb + s * IMG + oy * 28 + ox;
        float acc = 0.f;
        #pragma unroll
        for (int ky = 0; ky < 3; ++ky)
            #pragma unroll
            for (int kx = 0; kx < 3; ++kx)
                acc = fmaf(px[ky * 28 + kx], w[ky * 3 + kx], acc);
        h[s * HSTR + p] = acc;
    }
    __syncthreads();

    // ---- fc1: (32 x 676) @ (676 x 128)^T via V_WMMA_F32_16X16X4_F32 ----
    // 8 waves x one 16-col N tile; each wave drives TWO M tiles so every
    // B fragment (global/L2) feeds two wmma ops.
    const int wave = tid >> 5;
    const int lane = tid & 31;
    const int n0   = wave * 16;            // this wave's output-column base
    const int row  = lane & 15;            // A: M index / B: N index
    const int kk   = (lane >> 4) << 1;     // K sub-offset: 0 (lanes 0-15) or 2 (lanes 16-31)

    // A frags: h[m][k0+kk .. +1]  (contiguous -> ds_load_b64)
    const float* ap0 = h + row * HSTR + kk;            // samples 0..15
    const float* ap1 = h + (16 + row) * HSTR + kk;     // samples 16..31
    // B frag: B[k][n] = fc1_w[n][k]; row-major -> contiguous pair (global_load_b64)
    const float* bp  = fc1_w + (size_t)(n0 + row) * CONV + kk;

    v8f c0 = {}, c1 = {};
    #pragma unroll 4
    for (int k0 = 0; k0 < CONV; k0 += 4) {
        v2f b  = *(const v2f*)(bp  + k0);
        v2f a0 = *(const v2f*)(ap0 + k0);
        v2f a1 = *(const v2f*)(ap1 + k0);
        // (neg_a, A, neg_b, B, c_mod, C, reuse_a, reuse_b)
        c0 = __builtin_amdgcn_wmma_f32_16x16x4_f32(false, a0, false, b,
                                                   (short)0, c0, false, false);
        c1 = __builtin_amdgcn_wmma_f32_16x16x4_f32(false, a1, false, b,
                                                   (short)0, c1, false, false);
    }
    __syncthreads();   // all reads of h done; safe to overwrite with h2

    // bias + ReLU, scatter D tiles to LDS using documented C/D layout:
    // VGPR v: lanes 0-15 -> M=v, lanes 16-31 -> M=v+8; N = lane&15
    const float bias  = fc1_b[n0 + row];
    const int   mbase = (lane >> 4) << 3;
    #pragma unroll
    for (int v = 0; v < 8; ++v) {
        float v0 = c0[v] + bias;
        float v1 = c1[v] + bias;
        h2[(mbase + v)      * H1 + (n0 + row)] = v0 > 0.f ? v0 : 0.f;
        h2[(mbase + v + 16) * H1 + (n0 + row)] = v1 > 0.f ? v1 : 0.f;
    }
    __syncthreads();

    // ---------------- fc2: (32 x 128) @ (128 x 10)^T + bias ----------------
    for (int t = tid; t < TS * H2; t += 256) {
        int s = t / H2;
        int o = t - s * H2;
        const float* hv = h2 + s * H1;
        const float* wv = fc2_w + o * H1;
        float acc = fc2_b[o];
        #pragma unroll 8
        for (int k = 0; k < H1; ++k) acc = fmaf(hv[k], wv[k], acc);
        out[(size_t)(s0 + s) * H2 + o] = acc;
    }
}

extern "C" void kernel_launch(void* const* d_in, const int* in_sizes, int n_in,
                              void* d_out, int out_size, void* d_ws, size_t ws_size,
                              hipStream_t stream) {
    const float* x      = (const float*)d_in[0];
    const float* conv_w = (const float*)d_in[1];
    const float* fc1_w  = (const float*)d_in[2];
    const float* fc1_b  = (const float*)d_in[3];
    const float* fc2_w  = (const float*)d_in[4];
    const float* fc2_b  = (const float*)d_in[5];
    float* out = (float*)d_out;

    const int B = in_sizes[0] / IMG;     // 65536
    fused_digit_mlp<<<B / TS, 256, 0, stream>>>(x, conv_w, fc1_w, fc1_b,
                                                fc2_w, fc2_b, out);
}